// GCN_29446295781425
// MI455X (gfx1250) — compile-verified
//
#include <hip/hip_runtime.h>
#include <hip/hip_bf16.h>

// ---------------- problem constants (match reference) ----------------
#define NUM_GRAPHS      1024
#define NODES_PER_GRAPH 115
#define N_NODES         (NUM_GRAPHS * NODES_PER_GRAPH)   // 117760
#define N_EDGES         (N_NODES * 32)                   // 3768320
#define IN_FEAT         115
#define HIDDEN          64
#define HC2             256
#define GFEAT           (NODES_PER_GRAPH * HIDDEN)       // 7360

typedef __attribute__((ext_vector_type(2))) float v2f;
typedef __attribute__((ext_vector_type(8))) float v8f;

// ---------------------------------------------------------------------
// Kernel 1: deg[i] = 1.0 (self-loop weight)
__global__ __launch_bounds__(256) void k_init_deg(float* __restrict__ deg) {
    int i = blockIdx.x * 256 + threadIdx.x;
    if (i < N_NODES) deg[i] = 1.0f;
}

// Kernel 2: deg[dst[e]] += w[e]
__global__ __launch_bounds__(256) void k_deg_accum(const int* __restrict__ ei,
                                                   const float* __restrict__ w,
                                                   float* __restrict__ deg) {
    int e = blockIdx.x * 256 + threadIdx.x;
    if (e < N_EDGES) {
        int dst = ei[N_EDGES + e];          // edge_index[1]
        atomicAdd(&deg[dst], w[e]);
    }
}

// ---------------------------------------------------------------------
// Kernel 3: H = X @ W1  (fp32 WMMA 16x16x4), LDS-staged, branch-free K loop.
// Block: 256 threads = 8 waves; each wave owns a 16-row x 64-col tile.
// Block covers 128 rows. K padded 115 -> 116 with zeros (no guards, no OOB).
#define G1_STR 116                            // 115 + 1 zero pad; 116 mod 64 -> conflict-free
__global__ __launch_bounds__(256) void k_gemm1(const float* __restrict__ X,
                                               const float* __restrict__ W1,
                                               float* __restrict__ H) {
    __shared__ float sW1[HIDDEN * G1_STR];    // transposed: sW1[c*116 + k]  (29.7 KB)
    __shared__ float sX[128 * G1_STR];        // sX[r*116 + k]               (59.4 KB)
    const int tid  = threadIdx.x;
    const int lane = tid & 31;
    const int wave = tid >> 5;
    const int half = lane >> 4;               // 0 = lanes 0-15, 1 = lanes 16-31
    const int l15  = lane & 15;
    const int blockRow0 = blockIdx.x * 128;   // 920 blocks * 128 = 117760 rows

    // stage W1 transposed + zero-pad k=115
    for (int idx = tid; idx < IN_FEAT * HIDDEN; idx += 256) {
        int k = idx >> 6, c = idx & 63;
        sW1[c * G1_STR + k] = W1[idx];
    }
    if (tid < HIDDEN) sW1[tid * G1_STR + IN_FEAT] = 0.0f;

    // stage X tile (coalesced: X rows are contiguous) + zero-pad k=115
    const float* Xb = X + (long long)blockRow0 * IN_FEAT;
    for (int idx = tid; idx < 128 * IN_FEAT; idx += 256) {
        int r = idx / IN_FEAT;
        int k = idx - r * IN_FEAT;
        sX[r * G1_STR + k] = Xb[idx];
    }
    if (tid < 128) sX[tid * G1_STR + IN_FEAT] = 0.0f;
    __syncthreads();

    v8f acc[4] = {{}, {}, {}, {}};
    const float* ar = &sX[(wave * 16 + l15) * G1_STR];
    for (int k0 = 0; k0 < IN_FEAT; k0 += 4) { // 29 iters, k reaches 115 (zero pad)
        const int ka = k0 + half * 2;         // even -> 8B-aligned LDS b64 loads
        v2f a = *(const v2f*)(ar + ka);
#pragma unroll
        for (int nt = 0; nt < 4; ++nt) {
            v2f b = *(const v2f*)(&sW1[(nt * 16 + l15) * G1_STR + ka]);
            acc[nt] = __builtin_amdgcn_wmma_f32_16x16x4_f32(
                false, a, false, b, (short)0, acc[nt], false, false);
        }
    }
    // C/D layout: VGPR v -> row base + half*8 + v, col = nt*16 + l15
#pragma unroll
    for (int nt = 0; nt < 4; ++nt)
#pragma unroll
        for (int v = 0; v < 8; ++v) {
            const int row = blockRow0 + wave * 16 + half * 8 + v;
            const int col = nt * 16 + l15;
            H[row * HIDDEN + col] = acc[nt][v];
        }
}

// ---------------------------------------------------------------------
// Kernel 4: dis = rsqrt(deg); agg = h / deg  (self-loop contribution)
__global__ __launch_bounds__(256) void k_prep(const float* __restrict__ deg,
                                              const float* __restrict__ h,
                                              float* __restrict__ dis,
                                              float* __restrict__ agg) {
    int t = blockIdx.x * 256 + threadIdx.x;
    if (t < N_NODES * HIDDEN) {
        int node = t >> 6;
        float d = deg[node];
        agg[t] = h[t] / d;
        if ((t & 63) == 0) dis[node] = rsqrtf(d);
    }
}

// ---------------------------------------------------------------------
// Kernel 5: edge aggregation. 16 lanes x float4 = 64 features per edge.
// agg[dst, f] += dis[src]*w*dis[dst] * h[src, f]
__global__ __launch_bounds__(256) void k_edge_agg(const int* __restrict__ ei,
                                                  const float* __restrict__ w,
                                                  const float* __restrict__ dis,
                                                  const float* __restrict__ h,
                                                  float* __restrict__ agg) {
    long long t = (long long)blockIdx.x * 256 + threadIdx.x;
    int e = (int)(t >> 4);
    int q = (int)(t & 15);
    if (e < N_EDGES) {
        int src = ei[e];
        int dst = ei[N_EDGES + e];
        float norm = dis[src] * w[e] * dis[dst];
        int f0 = q * 4;
        const float4 hv = *(const float4*)(h + (long long)src * HIDDEN + f0);
        float* ap = agg + (long long)dst * HIDDEN + f0;
        atomicAdd(ap + 0, norm * hv.x);
        atomicAdd(ap + 1, norm * hv.y);
        atomicAdd(ap + 2, norm * hv.z);
        atomicAdd(ap + 3, norm * hv.w);
    }
}

// ---------------------------------------------------------------------
// Kernel 6: F = relu(agg + b1), in place (F viewed as [1024][7360])
__global__ __launch_bounds__(256) void k_relu_bias(float* __restrict__ agg,
                                                   const float* __restrict__ b1) {
    int t = blockIdx.x * 256 + threadIdx.x;
    if (t < N_NODES * HIDDEN) {
        float v = agg[t] + b1[t & 63];
        agg[t] = v > 0.0f ? v : 0.0f;
    }
}

// Kernel 7a: zero out2 (K-split partial accumulator)
__global__ __launch_bounds__(256) void k_zero(float* __restrict__ p, int n) {
    int t = blockIdx.x * 256 + threadIdx.x;
    if (t < n) p[t] = 0.0f;
}

// ---------------------------------------------------------------------
// Kernel 7: out2 += F @ W2 (fp32 WMMA, K-split x5, LDS-staged chunks of 32 K).
// Grid: 32 row-blocks x 5 K-segments = 160 blocks, 8 waves each.
// Block covers 32 rows x 256 cols; wave tile = 16 rows x 64 cols.
#define KSPLIT 5
#define KSEG   (GFEAT / KSPLIT)               // 1472 (multiple of 32)
#define KC     32
#define BSTR   36                             // pad: 16B-aligned rows, distinct banks
__global__ __launch_bounds__(256) void k_gemm2(const float* __restrict__ F,
                                               const float* __restrict__ W2,
                                               float* __restrict__ out2) {
    __shared__ float sB[HC2 * BSTR];          // W2^T chunk: sB[c*36 + kk]  (36.9 KB)
    __shared__ float sA[32 * BSTR];           // A chunk:    sA[r*36 + kk]  ( 4.6 KB)
    const int tid  = threadIdx.x;
    const int lane = tid & 31;
    const int wave = tid >> 5;
    const int half = lane >> 4;
    const int l15  = lane & 15;
    const int seg  = blockIdx.x >> 5;         // K segment 0..4
    const int b32  = blockIdx.x & 31;         // row block 0..31
    const int rowBase = b32 * 32;
    const int r0l  = (wave >> 2) * 16;        // local row tile: 0 or 16
    const int c0   = (wave & 3) * 64;         // col group
    const int kBeg = seg * KSEG;

    v8f acc[4] = {{}, {}, {}, {}};
    for (int k0 = kBeg; k0 < kBeg + KSEG; k0 += KC) {
        __syncthreads();
        // stage W2[k0:k0+32, 0:256] transposed; coalesced 1KB rows
#pragma unroll 4
        for (int i = 0; i < KC; ++i)
            sB[tid * BSTR + i] = W2[(long long)(k0 + i) * HC2 + tid];
        // stage F[rowBase:rowBase+32, k0:k0+32]; float4 per thread
        {
            int r = tid >> 3, j = (tid & 7) * 4;
            const float4 fv = *(const float4*)(F + (long long)(rowBase + r) * GFEAT + k0 + j);
            *(float4*)(&sA[r * BSTR + j]) = fv;
        }
        __syncthreads();
#pragma unroll
        for (int kk = 0; kk < KC; kk += 4) {
            const int ka = kk + half * 2;
            v2f a = *(const v2f*)(&sA[(r0l + l15) * BSTR + ka]);
#pragma unroll
            for (int nt = 0; nt < 4; ++nt) {
                v2f b = *(const v2f*)(&sB[(c0 + nt * 16 + l15) * BSTR + ka]);
                acc[nt] = __builtin_amdgcn_wmma_f32_16x16x4_f32(
                    false, a, false, b, (short)0, acc[nt], false, false);
            }
        }
    }
    // K-split accumulation
#pragma unroll
    for (int nt = 0; nt < 4; ++nt)
#pragma unroll
        for (int v = 0; v < 8; ++v) {
            const int row = rowBase + r0l + half * 8 + v;
            const int col = c0 + nt * 16 + l15;
            atomicAdd(&out2[row * HC2 + col], acc[nt][v]);
        }
}

// ---------------------------------------------------------------------
// Kernel 8: out[g] = dot(out2[g,:] + b2, W3) + b3  — one wave per graph
__global__ __launch_bounds__(256) void k_final(const float* __restrict__ out2,
                                               const float* __restrict__ b2,
                                               const float* __restrict__ W3,
                                               const float* __restrict__ b3,
                                               float* __restrict__ out) {
    int t    = blockIdx.x * 256 + threadIdx.x;
    int g    = t >> 5;
    int lane = t & 31;
    if (g < NUM_GRAPHS) {
        float s = 0.0f;
        for (int j = lane; j < HC2; j += 32)
            s += (out2[g * HC2 + j] + b2[j]) * W3[j];
#pragma unroll
        for (int off = 16; off > 0; off >>= 1)
            s += __shfl_xor(s, off, 32);
        if (lane == 0) out[g] = s + b3[0];
    }
}

// ---------------------------------------------------------------------
extern "C" void kernel_launch(void* const* d_in, const int* in_sizes, int n_in,
                              void* d_out, int out_size, void* d_ws, size_t ws_size,
                              hipStream_t stream) {
    const float* x  = (const float*)d_in[0];   // [N_NODES, IN_FEAT]
    const int*   ei = (const int*)  d_in[1];   // [2, N_EDGES]
    const float* ew = (const float*)d_in[2];   // [N_EDGES]
    // d_in[3] = batch (unused: equal-sized graphs)
    const float* W1 = (const float*)d_in[4];   // [IN_FEAT, HIDDEN]
    const float* b1 = (const float*)d_in[5];   // [HIDDEN]
    const float* W2 = (const float*)d_in[6];   // [GFEAT, HC2]
    const float* b2 = (const float*)d_in[7];   // [HC2]
    const float* W3 = (const float*)d_in[8];   // [HC2]
    const float* b3 = (const float*)d_in[9];   // [1]
    float* out = (float*)d_out;                // [NUM_GRAPHS]

    // workspace layout (floats), ~62.3 MB total
    float* ws   = (float*)d_ws;
    float* deg  = ws;                               // N_NODES
    float* dis  = deg + N_NODES;                    // N_NODES
    float* h    = dis + N_NODES;                    // N_NODES*HIDDEN
    float* agg  = h   + (size_t)N_NODES * HIDDEN;   // N_NODES*HIDDEN
    float* out2 = agg + (size_t)N_NODES * HIDDEN;   // NUM_GRAPHS*HC2

    k_init_deg <<<(N_NODES + 255) / 256, 256, 0, stream>>>(deg);
    k_deg_accum<<<(N_EDGES + 255) / 256, 256, 0, stream>>>(ei, ew, deg);
    k_gemm1    <<<N_NODES / 128, 256, 0, stream>>>(x, W1, h);
    k_prep     <<<(N_NODES * HIDDEN) / 256, 256, 0, stream>>>(deg, h, dis, agg);
    k_edge_agg <<<((long long)N_EDGES * 16) / 256, 256, 0, stream>>>(ei, ew, dis, h, agg);
    k_relu_bias<<<(N_NODES * HIDDEN) / 256, 256, 0, stream>>>(agg, b1);
    k_zero     <<<(NUM_GRAPHS * HC2) / 256, 256, 0, stream>>>(out2, NUM_GRAPHS * HC2);
    k_gemm2    <<<32 * KSPLIT, 256, 0, stream>>>(agg, W2, out2);
    k_final    <<<(NUM_GRAPHS * 32) / 256, 256, 0, stream>>>(out2, b2, W3, b3, out);
}